// GATLayer_16020228014948
// MI455X (gfx1250) — compile-verified
//
#include <hip/hip_runtime.h>

// ---------------- problem constants (match reference) ----------------
#define NN    50000
#define EE    600000
#define INF_  128     // n_in_feats
#define EFD   64      // n_edge_feats
#define HH    4       // heads
#define OO    32      // per-head dim
#define HO    128     // H*O
#define SLOPE 0.2f

typedef __attribute__((ext_vector_type(2))) float v2f;
typedef __attribute__((ext_vector_type(8))) float v8f;

// order-preserving float <-> uint map for atomicMax on floats
__device__ __forceinline__ unsigned fenc(float x) {
    unsigned u = __float_as_uint(x);
    return (u & 0x80000000u) ? ~u : (u | 0x80000000u);
}
__device__ __forceinline__ float fdec(unsigned u) {
    unsigned v = (u & 0x80000000u) ? (u & 0x7FFFFFFFu) : ~u;
    return __uint_as_float(v);
}
#define ENC_NEG_INF 0x007FFFFFu   // fenc(-inf)

// ---------------- init kernels ----------------
__global__ __launch_bounds__(256) void k_fill_f32(float* p, int n, float v) {
    int t = blockIdx.x * blockDim.x + threadIdx.x;
    if (t < n) p[t] = v;
}
__global__ __launch_bounds__(256) void k_fill_u32(unsigned* p, int n, unsigned v) {
    int t = blockIdx.x * blockDim.x + threadIdx.x;
    if (t < n) p[t] = v;
}

// ---------------- WMMA fp32 GEMMs ----------------
// feat = A @ W^T + b ; res = A @ Wres^T + bres     A:[NN,128], W/Wres:[128,128]
// one wave = one 16x16 tile; 8 waves/block cover all 8 column tiles of a row tile.
__global__ __launch_bounds__(256) void k_gemm_node(
    const float* __restrict__ A,
    const float* __restrict__ W,  const float* __restrict__ bb,
    const float* __restrict__ Wr, const float* __restrict__ br,
    float* __restrict__ feat, float* __restrict__ res)
{
    const int lane = threadIdx.x & 31;
    const int wv   = threadIdx.x >> 5;     // column tile 0..7
    const int row0 = blockIdx.x << 4;
    const int col0 = wv << 4;
    const int l15  = lane & 15;
    const int hi   = lane >> 4;            // lane half selects K pair {0,1} vs {2,3}
    const float* arow = A  + (size_t)(row0 + l15) * INF_;
    const float* wrow = W  + (size_t)(col0 + l15) * INF_;
    const float* rrow = Wr + (size_t)(col0 + l15) * INF_;
    v8f acc0 = {}; v8f acc1 = {};
    for (int k = 0; k < INF_; k += 4) {
        const int ka = k + hi * 2;
        v2f av = *(const v2f*)(arow + ka);     // A fragment: rows in lanes, K in VGPRs
        v2f wf = *(const v2f*)(wrow + ka);     // B fragment: B[k][n] = W[col][k]
        v2f rf = *(const v2f*)(rrow + ka);
        acc0 = __builtin_amdgcn_wmma_f32_16x16x4_f32(false, av, false, wf,
                                                     (short)0, acc0, false, false);
        acc1 = __builtin_amdgcn_wmma_f32_16x16x4_f32(false, av, false, rf,
                                                     (short)0, acc1, false, false);
    }
    const int oc = col0 + l15;
    const float b0 = bb[oc], b1 = br[oc];
    for (int r = 0; r < 8; ++r) {
        const int orow = row0 + r + hi * 8;    // C/D layout: VGPR r -> rows r, r+8
        feat[(size_t)orow * HO + oc] = acc0[r] + b0;
        res [(size_t)orow * HO + oc] = acc1[r] + b1;
    }
}

// ebuf = edge_feats @ We^T + be      edge_feats:[EE,64], We:[128,64]
__global__ __launch_bounds__(256) void k_gemm_edge(
    const float* __restrict__ A, const float* __restrict__ W,
    const float* __restrict__ bb, float* __restrict__ out)
{
    const int lane = threadIdx.x & 31;
    const int wv   = threadIdx.x >> 5;
    const int row0 = blockIdx.x << 4;
    const int col0 = wv << 4;
    const int l15  = lane & 15;
    const int hi   = lane >> 4;
    const float* arow = A + (size_t)(row0 + l15) * EFD;
    const float* wrow = W + (size_t)(col0 + l15) * EFD;
    v8f acc = {};
    for (int k = 0; k < EFD; k += 4) {
        const int ka = k + hi * 2;
        v2f av = *(const v2f*)(arow + ka);
        v2f wf = *(const v2f*)(wrow + ka);
        acc = __builtin_amdgcn_wmma_f32_16x16x4_f32(false, av, false, wf,
                                                    (short)0, acc, false, false);
    }
    const int oc = col0 + l15;
    const float b0 = bb[oc];
    for (int r = 0; r < 8; ++r) {
        const int orow = row0 + r + hi * 8;
        out[(size_t)orow * HO + oc] = acc[r] + b0;
    }
}

// ---------------- phase 1: GATv2 attention ----------------
// one wave per edge: score[e,h] = sum_o lrelu(feat[src]+feat[dst]) * attn ; atomic segment max
__global__ __launch_bounds__(256) void k_edge_score(
    const float* __restrict__ feat, const int* __restrict__ src, const int* __restrict__ dst,
    const float* __restrict__ attn, float* __restrict__ score, unsigned* __restrict__ smax)
{
    const int lane = threadIdx.x & 31;
    const int e = blockIdx.x * (blockDim.x >> 5) + (threadIdx.x >> 5);
    if (e >= EE) return;
    const int s = src[e], d = dst[e];
    const float4 fs = reinterpret_cast<const float4*>(feat + (size_t)s * HO)[lane];
    const float4 fd = reinterpret_cast<const float4*>(feat + (size_t)d * HO)[lane];
    const float4 av = reinterpret_cast<const float4*>(attn)[lane];
    float p = 0.f;
    {
        float x;
        x = fs.x + fd.x; p += (x > 0.f ? x : SLOPE * x) * av.x;
        x = fs.y + fd.y; p += (x > 0.f ? x : SLOPE * x) * av.y;
        x = fs.z + fd.z; p += (x > 0.f ? x : SLOPE * x) * av.z;
        x = fs.w + fd.w; p += (x > 0.f ? x : SLOPE * x) * av.w;
    }
    // reduce across the 8 lanes of one head (lanes 8h..8h+7)
    p += __shfl_xor(p, 1, 32);
    p += __shfl_xor(p, 2, 32);
    p += __shfl_xor(p, 4, 32);
    if ((lane & 7) == 0) {
        const int h = lane >> 3;
        score[(size_t)e * HH + h] = p;
        atomicMax(smax + (size_t)d * HH + h, fenc(p));
    }
}

// per (edge,head): ex = exp(score - smax[dst]); denom += ex ; score <- ex (unnormalized alpha)
__global__ __launch_bounds__(256) void k_edge_exp(
    const int* __restrict__ dst, float* __restrict__ score,
    const unsigned* __restrict__ smax, float* __restrict__ denom)
{
    int t = blockIdx.x * blockDim.x + threadIdx.x;
    if (t >= EE * HH) return;
    const int e = t >> 2, h = t & 3;
    const int d = dst[e];
    float mx = fdec(smax[(size_t)d * HH + h]);          // finite: this edge contributed
    float ex = expf(score[t] - mx);
    score[t] = ex;
    atomicAdd(denom + (size_t)d * HH + h, ex);
}

// per (edge,channel): rst[dst,c] += ex * feat[src,c]   (normalize by denom at node)
__global__ __launch_bounds__(256) void k_edge_aggregate(
    const int* __restrict__ src, const int* __restrict__ dst,
    const float* __restrict__ score, const float* __restrict__ feat,
    float* __restrict__ rst)
{
    int t = blockIdx.x * blockDim.x + threadIdx.x;
    if (t >= EE * HO) return;
    const int e = t >> 7, c = t & 127, h = c >> 5;
    const int s = src[e], d = dst[e];
    const float ex = score[(size_t)e * HH + h];
    atomicAdd(rst + (size_t)d * HO + c, ex * feat[(size_t)s * HO + c]);
}

// h = relu(rst/denom + res)   (written in place over res)
__global__ __launch_bounds__(256) void k_node_combine(
    const float* __restrict__ rst, const float* __restrict__ denom,
    float* __restrict__ res_h)
{
    int t = blockIdx.x * blockDim.x + threadIdx.x;
    if (t >= NN * HO) return;
    const int n = t >> 7, c = t & 127, h = c >> 5;
    const float den = denom[(size_t)n * HH + h];
    float v = (den > 0.f) ? (rst[t] / den) : 0.f;
    v += res_h[t];
    res_h[t] = v > 0.f ? v : 0.f;
}

// ---------------- phase 2: mailbox softmax over (dst, h, o) ----------------
__global__ __launch_bounds__(256) void k_msg_max(
    const int* __restrict__ src, const int* __restrict__ dst,
    const float* __restrict__ hb, const float* __restrict__ ebuf,
    unsigned* __restrict__ mmax)
{
    int t = blockIdx.x * blockDim.x + threadIdx.x;
    if (t >= EE * HO) return;
    const int e = t >> 7, c = t & 127;
    const int s = src[e], d = dst[e];
    const float m = hb[(size_t)s * HO + c] + ebuf[t];
    atomicMax(mmax + (size_t)d * HO + c, fenc(m));
}

__global__ __launch_bounds__(256) void k_msg_sum(
    const int* __restrict__ src, const int* __restrict__ dst,
    const float* __restrict__ hb, const float* __restrict__ ebuf,
    const unsigned* __restrict__ mmax, float* __restrict__ den2,
    float* __restrict__ numer)
{
    int t = blockIdx.x * blockDim.x + threadIdx.x;
    if (t >= EE * HO) return;
    const int e = t >> 7, c = t & 127;
    const int s = src[e], d = dst[e];
    const float m  = hb[(size_t)s * HO + c] + ebuf[t];
    const float mm = fdec(mmax[(size_t)d * HO + c]);    // finite for touched channels
    const float ex = expf(m - mm);
    atomicAdd(den2  + (size_t)d * HO + c, ex);
    atomicAdd(numer + (size_t)d * HO + c, ex * m);
}

// out = den2 > 0 ? numer/den2 : 0   (numer accumulated in d_out, finalized in place)
__global__ __launch_bounds__(256) void k_finalize(
    const float* __restrict__ den2, float* __restrict__ out)
{
    int t = blockIdx.x * blockDim.x + threadIdx.x;
    if (t >= NN * HO) return;
    const float dn = den2[t];
    out[t] = (dn > 0.f) ? (out[t] / dn) : 0.f;
}

// ---------------- host launcher ----------------
static inline int cdiv(long long a, int b) { return (int)((a + b - 1) / b); }

extern "C" void kernel_launch(void* const* d_in, const int* in_sizes, int n_in,
                              void* d_out, int out_size, void* d_ws, size_t ws_size,
                              hipStream_t stream) {
    (void)in_sizes; (void)n_in; (void)out_size; (void)ws_size;
    const float* node  = (const float*)d_in[0];
    const float* edgef = (const float*)d_in[1];
    const int*   src   = (const int*)  d_in[2];
    const int*   dst   = (const int*)  d_in[3];
    const float* W     = (const float*)d_in[4];
    const float* b     = (const float*)d_in[5];
    const float* attn  = (const float*)d_in[6];
    const float* Wres  = (const float*)d_in[7];
    const float* bres  = (const float*)d_in[8];
    const float* We    = (const float*)d_in[9];
    const float* be    = (const float*)d_in[10];
    float* out = (float*)d_out;

    // workspace carve-up (256B aligned)
    char* base = (char*)d_ws;
    size_t off = 0;
    auto carve = [&](size_t bytes) -> char* {
        char* p = base + off;
        off += (bytes + 255) & ~(size_t)255;
        return p;
    };
    float*    feat  = (float*)   carve((size_t)NN * HO * 4);
    float*    res_h = (float*)   carve((size_t)NN * HO * 4);   // res, then h in place
    float*    ebuf  = (float*)   carve((size_t)EE * HO * 4);
    float*    score = (float*)   carve((size_t)EE * HH * 4);
    unsigned* smax  = (unsigned*)carve((size_t)NN * HH * 4);
    float*    den1  = (float*)   carve((size_t)NN * HH * 4);
    float*    rst   = (float*)   carve((size_t)NN * HO * 4);
    unsigned* mmax  = (unsigned*)carve((size_t)NN * HO * 4);
    float*    den2  = (float*)   carve((size_t)NN * HO * 4);

    const int B = 256;
    // --- init accumulators (every call: deterministic) ---
    k_fill_u32<<<cdiv(NN * HH, B), B, 0, stream>>>(smax, NN * HH, ENC_NEG_INF);
    k_fill_f32<<<cdiv(NN * HH, B), B, 0, stream>>>(den1, NN * HH, 0.f);
    k_fill_f32<<<cdiv((long long)NN * HO, B), B, 0, stream>>>(rst,  NN * HO, 0.f);
    k_fill_u32<<<cdiv((long long)NN * HO, B), B, 0, stream>>>(mmax, NN * HO, ENC_NEG_INF);
    k_fill_f32<<<cdiv((long long)NN * HO, B), B, 0, stream>>>(den2, NN * HO, 0.f);
    k_fill_f32<<<cdiv((long long)NN * HO, B), B, 0, stream>>>(out,  NN * HO, 0.f);

    // --- GEMMs (WMMA fp32) ---
    k_gemm_node<<<NN / 16, 256, 0, stream>>>(node, W, b, Wres, bres, feat, res_h);
    k_gemm_edge<<<EE / 16, 256, 0, stream>>>(edgef, We, be, ebuf);

    // --- phase 1: GATv2 attention + aggregation ---
    k_edge_score<<<cdiv(EE, 8), 256, 0, stream>>>(feat, src, dst, attn, score, smax);
    k_edge_exp<<<cdiv((long long)EE * HH, B), B, 0, stream>>>(dst, score, smax, den1);
    k_edge_aggregate<<<cdiv((long long)EE * HO, B), B, 0, stream>>>(src, dst, score, feat, rst);
    k_node_combine<<<cdiv((long long)NN * HO, B), B, 0, stream>>>(rst, den1, res_h);

    // --- phase 2: per-channel mailbox softmax reduce ---
    k_msg_max<<<cdiv((long long)EE * HO, B), B, 0, stream>>>(src, dst, res_h, ebuf, mmax);
    k_msg_sum<<<cdiv((long long)EE * HO, B), B, 0, stream>>>(src, dst, res_h, ebuf, mmax, den2, out);
    k_finalize<<<cdiv((long long)NN * HO, B), B, 0, stream>>>(den2, out);
}